// AttnDecoderRNN_12790412607796
// MI455X (gfx1250) — compile-verified
//
#include <hip/hip_runtime.h>
#include <hip/hip_bf16.h>
#include <math.h>

typedef float v2f __attribute__((ext_vector_type(2)));
typedef float v8f __attribute__((ext_vector_type(8)));

#define B_ 64
#define S_ 48
#define H_ 512
#define E_ 256
#define V_ 32000
#define T_ 31

// ---------------------------------------------------------------------------
// GEMM C = A @ W^T + bias, via V_WMMA_F32_16X16X4_F32.
// A row-major [M,LDA], W row-major [N,LDB]; LDA/LDB/K compile-time so all
// loads use immediate offsets from two base pointers (clause-friendly).
// Each wave computes a 64x16 strip of C: 4 accumulators along M sharing the
// B-operand registers each K-step (4x register reuse of W).
//   A layout (16x4 f32): lane = 16*half + m, {v0,v1} = A[m][k0+2*half + {0,1}]
//   B layout (4x16 f32): lane = 16*half + n, {v0,v1} = W[n][k0+2*half + {0,1}]
// Per 16-float K-chunk: phase 1 loads 4 B + 16 A float2s, phase 2 issues 16
// WMMAs (kk outer, mt inner: consecutive WMMAs use different accumulators,
// avoiding back-to-back D->A/B hazards).
// block = 128 threads (4 waves, wave w -> n-tile w); grid = (N/64, M/64).
// ---------------------------------------------------------------------------
template <int LDA, int LDB, int KTOT>
__global__ void k_gemm_nt_bias(const float* __restrict__ A,
                               const float* __restrict__ W,
                               const float* __restrict__ bias,
                               float* __restrict__ C, int ldc) {
  const int wave = threadIdx.x >> 5;
  const int n0 = blockIdx.x * 64 + wave * 16;
  const int m0 = blockIdx.y * 64;
  const int lane = threadIdx.x & 31;
  const int half = lane >> 4;   // 0: K pair {0,1}, 1: K pair {2,3}
  const int idx  = lane & 15;   // m (within tile) for A, n for B

  const float* ap = A + (size_t)(m0 + idx) * LDA + 2 * half;
  const float* bp = W + (size_t)(n0 + idx) * LDB + 2 * half;

  v8f acc[4];
#pragma unroll
  for (int mt = 0; mt < 4; ++mt)
    acc[mt] = (v8f){0.f, 0.f, 0.f, 0.f, 0.f, 0.f, 0.f, 0.f};

  for (int k = 0; k < KTOT; k += 16) {
    v2f bb[4];
    v2f aa[4][4];
    // ---- phase 1: all loads of this chunk (immediate offsets) ----
#pragma unroll
    for (int kk = 0; kk < 4; ++kk)
      bb[kk] = *(const v2f*)(bp + k + 4 * kk);
#pragma unroll
    for (int mt = 0; mt < 4; ++mt)
#pragma unroll
      for (int kk = 0; kk < 4; ++kk)
        aa[mt][kk] = *(const v2f*)(ap + mt * (16 * LDA) + k + 4 * kk);
    // ---- phase 2: 16 WMMAs ----
#pragma unroll
    for (int kk = 0; kk < 4; ++kk)
#pragma unroll
      for (int mt = 0; mt < 4; ++mt)
        acc[mt] = __builtin_amdgcn_wmma_f32_16x16x4_f32(
            false, aa[mt][kk], false, bb[kk], (short)0, acc[mt], false, false);
  }

  const int n = n0 + idx;
  const float bn = bias[n];
#pragma unroll
  for (int mt = 0; mt < 4; ++mt) {
#pragma unroll
    for (int v = 0; v < 8; ++v) {
      const int m = m0 + mt * 16 + v + 8 * half;  // C/D layout: vgpr v, lanes>=16 -> M+8
      C[(size_t)m * ldc + n] = acc[mt][v] + bn;
    }
  }
}

// ---------------------------------------------------------------------------
// Attention: scores -> softmax -> ctx, plus embedding gather into xbuf.
// One block per batch row b; 256 threads = 8 waves.
// ---------------------------------------------------------------------------
__global__ void k_attn(const float* __restrict__ q,        // [B,H]
                       const float* __restrict__ Uk,       // [B,S,H]
                       const float* __restrict__ keys,     // [B,S,H]
                       const float* __restrict__ Va,       // [H]
                       const float* __restrict__ bv,       // [1]
                       const int*   __restrict__ target,   // [B,T]
                       const float* __restrict__ emb,      // [V,E]
                       float* __restrict__ xbuf,           // [B, E+H]
                       float* __restrict__ attn_out,       // [B,T,S]
                       int t) {
  const int b = blockIdx.x;
  const int tid = threadIdx.x;
  const int wv = tid >> 5, lane = tid & 31;
  __shared__ float sc[S_];

  const float* qb  = q + b * H_;
  const float* Ukb = Uk + (size_t)b * S_ * H_;
  const float* kb  = keys + (size_t)b * S_ * H_;

  // scores[s] = bv + Va . tanh(q_b + Uk[b,s])   (wave wv handles s = wv, wv+8, ...)
  for (int s = wv; s < S_; s += 8) {
    float p = 0.f;
    for (int h = lane; h < H_; h += 32)
      p += Va[h] * tanhf(qb[h] + Ukb[s * H_ + h]);
    for (int off = 16; off > 0; off >>= 1)
      p += __shfl_down(p, off, 32);
    if (lane == 0) sc[s] = p + bv[0];
  }
  __syncthreads();

  // softmax over S (redundant per-thread; S=48 is tiny)
  float m = -INFINITY;
  for (int s = 0; s < S_; ++s) m = fmaxf(m, sc[s]);
  float sum = 0.f;
  for (int s = 0; s < S_; ++s) sum += expf(sc[s] - m);
  const float inv = 1.f / sum;
  __syncthreads();
  if (tid < S_) {
    const float w = expf(sc[tid] - m) * inv;
    sc[tid] = w;
    attn_out[((size_t)b * T_ + t) * S_ + tid] = w;
  }
  __syncthreads();

  // ctx[h] = sum_s w_s * keys[b,s,h]  -> xbuf[b, E + h]
  for (int h = tid; h < H_; h += 256) {
    float c = 0.f;
#pragma unroll 8
    for (int s = 0; s < S_; ++s) c += sc[s] * kb[s * H_ + h];
    xbuf[(size_t)b * (E_ + H_) + E_ + h] = c;
  }
  // embedding gather -> xbuf[b, 0:E]   (E == blockDim.x)
  const int tok = (t == 0) ? 0 : target[b * T_ + (t - 1)];
  xbuf[(size_t)b * (E_ + H_) + tid] = emb[(size_t)tok * E_ + tid];
}

// GRU gate fuse: r,z,n -> h_new   (PyTorch gate order r,z,n)
__global__ void k_gate(const float* __restrict__ gi, const float* __restrict__ gh,
                       const float* __restrict__ h, float* __restrict__ hnew, int n) {
  const int i = blockIdx.x * blockDim.x + threadIdx.x;
  if (i >= n) return;
  const int b = i / H_, j = i % H_;
  const float* gib = gi + (size_t)b * 3 * H_;
  const float* ghb = gh + (size_t)b * 3 * H_;
  const float r = 1.f / (1.f + expf(-(gib[j] + ghb[j])));
  const float z = 1.f / (1.f + expf(-(gib[H_ + j] + ghb[H_ + j])));
  const float nn = tanhf(gib[2 * H_ + j] + r * ghb[2 * H_ + j]);
  hnew[i] = (1.f - z) * nn + z * h[i];
}

// In-place log_softmax over one row of V elements per block (256 threads).
__global__ void k_logsoftmax(float* __restrict__ out, int V) {
  float* row = out + (size_t)blockIdx.x * V;
  const int tid = threadIdx.x, lane = tid & 31, wv = tid >> 5;
  __shared__ float sred[8];

  float m = -INFINITY;
  for (int i = tid; i < V; i += 256) m = fmaxf(m, row[i]);
  for (int off = 16; off > 0; off >>= 1) m = fmaxf(m, __shfl_down(m, off, 32));
  if (lane == 0) sred[wv] = m;
  __syncthreads();
  m = -INFINITY;
  for (int w = 0; w < 8; ++w) m = fmaxf(m, sred[w]);
  __syncthreads();

  float s = 0.f;
  for (int i = tid; i < V; i += 256) s += expf(row[i] - m);
  for (int off = 16; off > 0; off >>= 1) s += __shfl_down(s, off, 32);
  if (lane == 0) sred[wv] = s;
  __syncthreads();
  s = 0.f;
  for (int w = 0; w < 8; ++w) s += sred[w];
  const float lse = m + logf(s);

  for (int i = tid; i < V; i += 256) row[i] -= lse;
}

__global__ void k_copy(const float* __restrict__ src, float* __restrict__ dst, int n) {
  const int i = blockIdx.x * blockDim.x + threadIdx.x;
  if (i < n) dst[i] = src[i];
}

// ---------------------------------------------------------------------------
extern "C" void kernel_launch(void* const* d_in, const int* in_sizes, int n_in,
                              void* d_out, int out_size, void* d_ws, size_t ws_size,
                              hipStream_t stream) {
  const float* enc_out = (const float*)d_in[0];   // [B,S,H]
  const float* enc_hid = (const float*)d_in[1];   // [1,B,H]
  const int*   target  = (const int*)  d_in[2];   // [B,T]
  const float* emb     = (const float*)d_in[3];   // [V,E]
  const float* Wa      = (const float*)d_in[4];   // [H,H]
  const float* ba      = (const float*)d_in[5];   // [H]
  const float* Ua      = (const float*)d_in[6];   // [H,H]
  const float* bu      = (const float*)d_in[7];   // [H]
  const float* Va      = (const float*)d_in[8];   // [H]
  const float* bv      = (const float*)d_in[9];   // [1]
  const float* W_ih    = (const float*)d_in[10];  // [3H, E+H]
  const float* W_hh    = (const float*)d_in[11];  // [3H, H]
  const float* b_ih    = (const float*)d_in[12];  // [3H]
  const float* b_hh    = (const float*)d_in[13];  // [3H]
  const float* W_out   = (const float*)d_in[14];  // [V,H]
  const float* b_out   = (const float*)d_in[15];  // [V]

  // workspace layout (floats): ~7.7 MB total
  float* ws   = (float*)d_ws;
  float* Uk   = ws;                          // B*S*H = 1572864
  float* q    = Uk + (size_t)B_ * S_ * H_;   // B*H
  float* xbuf = q + B_ * H_;                 // B*(E+H)
  float* gi   = xbuf + B_ * (E_ + H_);       // B*3H
  float* gh   = gi + B_ * 3 * H_;            // B*3H
  float* hA   = gh + B_ * 3 * H_;            // B*H
  float* hB   = hA + B_ * H_;                // B*H

  float* out        = (float*)d_out;
  float* out_logits = out;                                   // [B,T,V]
  float* out_hidden = out + (size_t)B_ * T_ * V_;            // [1,B,H]
  float* out_attn   = out_hidden + (size_t)B_ * H_;          // [B,T,S]

  // Uk = keys @ Ua^T + bu   (M = B*S = 3072, N = H, K = H)
  k_gemm_nt_bias<H_, H_, H_><<<dim3(H_ / 64, (B_ * S_) / 64), 128, 0, stream>>>(
      enc_out, Ua, bu, Uk, H_);

  // h0
  k_copy<<<(B_ * H_ + 255) / 256, 256, 0, stream>>>(enc_hid, hA, B_ * H_);

  float* hc = hA;
  float* hn = hB;
  for (int t = 0; t < T_; ++t) {
    // q = h @ Wa^T + ba
    k_gemm_nt_bias<H_, H_, H_><<<dim3(H_ / 64, 1), 128, 0, stream>>>(
        hc, Wa, ba, q, H_);
    // attention + ctx + embedding gather
    k_attn<<<B_, 256, 0, stream>>>(q, Uk, enc_out, Va, bv, target, emb, xbuf,
                                   out_attn, t);
    // gi = [emb,ctx] @ W_ih^T + b_ih   (K = E+H = 768)
    k_gemm_nt_bias<E_ + H_, E_ + H_, E_ + H_>
        <<<dim3((3 * H_) / 64, 1), 128, 0, stream>>>(
            xbuf, W_ih, b_ih, gi, 3 * H_);
    // gh = h @ W_hh^T + b_hh
    k_gemm_nt_bias<H_, H_, H_><<<dim3((3 * H_) / 64, 1), 128, 0, stream>>>(
        hc, W_hh, b_hh, gh, 3 * H_);
    // h_new
    k_gate<<<(B_ * H_ + 255) / 256, 256, 0, stream>>>(gi, gh, hc, hn, B_ * H_);
    // logits -> d_out[b, t, :]   (ldc = T*V, column offset t*V)
    k_gemm_nt_bias<H_, H_, H_><<<dim3(V_ / 64, 1), 128, 0, stream>>>(
        hn, W_out, b_out, out_logits + (size_t)t * V_, T_ * V_);
    float* tmp = hc; hc = hn; hn = tmp;
  }

  // log_softmax over every [b,t] row
  k_logsoftmax<<<B_ * T_, 256, 0, stream>>>(out_logits, V_);
  // final hidden
  k_copy<<<(B_ * H_ + 255) / 256, 256, 0, stream>>>(hc, out_hidden, B_ * H_);
}